// TripletLoss_WRT_62173946577475
// MI455X (gfx1250) — compile-verified
//
#include <hip/hip_runtime.h>

typedef __attribute__((ext_vector_type(2))) float    v2f;
typedef __attribute__((ext_vector_type(8))) float    v8f;
typedef __attribute__((ext_vector_type(4))) unsigned v4u;
typedef __attribute__((ext_vector_type(4))) int      v4i;
typedef __attribute__((ext_vector_type(8))) int      v8i;

#define N_ROWS 8192
#define DIM 512
#define LDS_STRIDE 516   // 512 + 2x2 dword pads (every 256 dwords) -> conflict-free b64 frags
#define BLK_ROWS 64
#define CHUNK 32
#define THREADS 128      // 4 waves, one 16-row strip per wave

// ---------------- Tensor Data Mover: 2D tile (512 x rows) global -> LDS ----------------
// D# per CDNA5 ISA ch.8: group0 = {flags, lds_addr, global_addr, type=2},
// group1 = {data_size=4B | pad_enable | pad_interval=256dw | pad_amount=2dw,
//           tensor_dim0=512, tensor_dim1=rows, tile_dim0=512, tile_dim1=rows,
//           tensor_dim0_stride=512}.  Padding of 2 dwords every 256 dwords gives the
// 516-dword row footprint the compute side expects.
__device__ __forceinline__ void tdm_load_tile(unsigned ldsOff, const float* gptr, int rows) {
  unsigned long long ga = (unsigned long long)(size_t)gptr;
  v4u g0;
  g0[0] = 1u;                                               // count=1, user descriptor
  g0[1] = ldsOff;                                           // lds_addr (bytes)
  g0[2] = (unsigned)ga;                                     // global_addr[31:0]
  g0[3] = (unsigned)((ga >> 32) & 0x01FFFFFFu) | (2u << 30);// global_addr[56:32] | type=2
  v8i g1;
  g1[0] = (int)((2u << 16) | (1u << 20) | (7u << 22) | (1u << 25)); // 4B, pad 2dw/256dw
  g1[1] = (int)(512u << 16);                                // tensor_dim0 = 512
  g1[2] = (int)((unsigned)rows << 16);                      // tensor_dim1 = rows
  g1[3] = (int)(512u << 16);                                // tile_dim0 = 512
  g1[4] = (int)rows;                                        // tile_dim1 = rows
  g1[5] = 512;                                              // tensor_dim0_stride = 512
  g1[6] = 0; g1[7] = 0;
  v4i gz = {0, 0, 0, 0};
#if __clang_major__ >= 23
  v8i gz8 = {0, 0, 0, 0, 0, 0, 0, 0};
  __builtin_amdgcn_tensor_load_to_lds(g0, g1, gz, gz, gz8, 0);
#else
  __builtin_amdgcn_tensor_load_to_lds(g0, g1, gz, gz, 0);
#endif
}

// ---------------- kernel 1: row squared norms (one wave per row) ----------------
__global__ void __launch_bounds__(256)
sqnorm_kernel(const float* __restrict__ X, float* __restrict__ sq) {
  const int lane = threadIdx.x & 31;
  const int wave = threadIdx.x >> 5;
  const int row  = blockIdx.x * 8 + wave;
  const float4* xr = (const float4*)(X + (size_t)row * DIM);
  float acc = 0.f;
#pragma unroll
  for (int i = 0; i < 4; ++i) {
    float4 v = xr[lane + i * 32];
    acc += v.x * v.x + v.y * v.y + v.z * v.z + v.w * v.w;
  }
#pragma unroll
  for (int m = 1; m < 32; m <<= 1) acc += __shfl_xor(acc, m, 32);
  if (lane == 0) sq[row] = acc;
}

// ------- kernel 2: fused Gram (fp32 WMMA) + masked soft-weighted reductions --------
// Each wave owns a full 16-row strip (rows never shared across waves), walks all 8192
// columns in 32-col chunks. B chunks arrive via double-buffered TDM DMA.
__global__ void __launch_bounds__(THREADS)
triplet_main_kernel(const float* __restrict__ X, const int* __restrict__ tgt,
                    const float* __restrict__ sq, float* __restrict__ rowloss) {
  extern __shared__ float smem[];
  float* ldsA = smem;                               // BLK_ROWS * LDS_STRIDE
  float* bufB0 = ldsA + BLK_ROWS * LDS_STRIDE;      // CHUNK * LDS_STRIDE
  float* bufB1 = bufB0 + CHUNK * LDS_STRIDE;        // CHUNK * LDS_STRIDE

  const int tid  = threadIdx.x;
  const int lane = tid & 31;
  const int wave = tid >> 5;
  const int half = lane >> 4;     // lane-half -> M offset 0 / +8, K offset 0 / +2
  const int ln   = lane & 15;     // N index within a 16-wide tile
  const int rowBase = blockIdx.x * BLK_ROWS;

  // Prologue: DMA the A tile (64x512) and the first B chunk; one wave issues.
  if (wave == 0) {
    tdm_load_tile((unsigned)(size_t)ldsA, X + (size_t)rowBase * DIM, BLK_ROWS);
    tdm_load_tile((unsigned)(size_t)bufB0, X, CHUNK);
  }

  // per-lane row scalars (the 8 rows this lane's accumulator VGPRs map to)
  float sqrow[8]; int trow[8];
#pragma unroll
  for (int r = 0; r < 8; ++r) {
    int rg = rowBase + wave * 16 + half * 8 + r;
    sqrow[r] = sq[rg];
    trow[r]  = tgt[rg];
  }

  // per-lane state: raw sums + plain max (exp(d) <= e^~50 fits fp32; masked zeros make
  // max(dist*mask) >= 0 and max(-dist*mask) == 0, so init 0 is exact).
  float Mp[8], Sp[8], Tp[8], Mn[8], Sn[8], Tn[8];
#pragma unroll
  for (int r = 0; r < 8; ++r) { Mp[r] = 0.f; Sp[r] = 0.f; Tp[r] = 0.f;
                                Mn[r] = 0.f; Sn[r] = 0.f; Tn[r] = 0.f; }

  const int aBase  = (wave * 16 + ln) * LDS_STRIDE + half * 2;
  const int bBase0 = (ln)      * LDS_STRIDE + half * 2;
  const int bBase1 = (16 + ln) * LDS_STRIDE + half * 2;

  for (int cb = 0; cb < N_ROWS; cb += CHUNK) {
    const int it = cb / CHUNK;
    float* bb = (it & 1) ? bufB1 : bufB0;

    if (wave == 0) __builtin_amdgcn_s_wait_tensorcnt(0);  // current buffer DMA done
    __syncthreads();                                      // publish LDS; all waves past
                                                          // last use of the other buffer
    if (wave == 0 && cb + CHUNK < N_ROWS) {
      float* nb = (it & 1) ? bufB0 : bufB1;
      tdm_load_tile((unsigned)(size_t)nb, X + (size_t)(cb + CHUNK) * DIM, CHUNK);
    }

    // column scalars for this chunk (small L2-resident arrays, per-lane gathers)
    float sqc0 = sq[cb + ln],        sqc1 = sq[cb + 16 + ln];
    int   tc0  = tgt[cb + ln],       tc1  = tgt[cb + 16 + ln];

    // ---- 16x32 tile per wave via fp32 WMMA; K split at the 256-dword mid-pad ----
    v8f acc0 = {};
    v8f acc1 = {};
#pragma unroll 8
    for (int ks = 0; ks < 64; ++ks) {               // K = 0..255
      v2f a  = *(const v2f*)(ldsA + aBase  + 4 * ks);
      v2f b0 = *(const v2f*)(bb   + bBase0 + 4 * ks);
      v2f b1 = *(const v2f*)(bb   + bBase1 + 4 * ks);
      acc0 = __builtin_amdgcn_wmma_f32_16x16x4_f32(false, a, false, b0,
                                                   (short)0, acc0, false, false);
      acc1 = __builtin_amdgcn_wmma_f32_16x16x4_f32(false, a, false, b1,
                                                   (short)0, acc1, false, false);
    }
#pragma unroll 8
    for (int ks = 64; ks < 128; ++ks) {             // K = 256..511 (+2 dword mid-pad)
      v2f a  = *(const v2f*)(ldsA + aBase  + 4 * ks + 2);
      v2f b0 = *(const v2f*)(bb   + bBase0 + 4 * ks + 2);
      v2f b1 = *(const v2f*)(bb   + bBase1 + 4 * ks + 2);
      acc0 = __builtin_amdgcn_wmma_f32_16x16x4_f32(false, a, false, b0,
                                                   (short)0, acc0, false, false);
      acc1 = __builtin_amdgcn_wmma_f32_16x16x4_f32(false, a, false, b1,
                                                   (short)0, acc1, false, false);
    }

    // ---- fused epilogue: distance + masked exp accumulations ----
#pragma unroll
    for (int s = 0; s < 2; ++s) {
      float sqc = (s == 0) ? sqc0 : sqc1;
      int   tc  = (s == 0) ? tc0  : tc1;
#pragma unroll
      for (int r = 0; r < 8; ++r) {
        float g  = (s == 0) ? acc0[r] : acc1[r];
        float d2 = sqrow[r] + sqc - 2.0f * g;
        float d  = sqrtf(fmaxf(d2, 1e-12f));
        bool pos = (trow[r] == tc);
        float ep = __expf(d);
        float en = __expf(-d);
        Mp[r] = fmaxf(Mp[r], pos ? d : 0.0f);
        Mn[r] = fmaxf(Mn[r], pos ? 0.0f : -d);
        Sp[r] += pos ? ep     : 0.0f;
        Tp[r] += pos ? d * ep : 0.0f;
        Sn[r] += pos ? 0.0f   : en;
        Tn[r] += pos ? 0.0f   : d * en;
      }
    }
  }

  // ---- reduce across the 16 lanes of each half (xor masks stay inside halves) ----
#pragma unroll
  for (int r = 0; r < 8; ++r) {
#pragma unroll
    for (int m = 1; m < 16; m <<= 1) {
      Sp[r] += __shfl_xor(Sp[r], m, 32);
      Tp[r] += __shfl_xor(Tp[r], m, 32);
      Sn[r] += __shfl_xor(Sn[r], m, 32);
      Tn[r] += __shfl_xor(Tn[r], m, 32);
      Mp[r]  = fmaxf(Mp[r], __shfl_xor(Mp[r], m, 32));
      Mn[r]  = fmaxf(Mn[r], __shfl_xor(Mn[r], m, 32));
    }
  }

  // rows are wave-private: finish the loss right here (lanes 0 and 16 hold results)
  if (ln == 0) {
#pragma unroll
    for (int r = 0; r < 8; ++r) {
      // exact eps handling: T/(S + eps*e^M) == sum(d*e^{d-M}) / (sum(e^{d-M}) + eps)
      float fp = Tp[r] / (Sp[r] + 1e-6f * __expf(Mp[r]));
      float cn = Tn[r] / (Sn[r] + 1e-6f * __expf(Mn[r]));
      float x = fp - cn;
      float loss = fmaxf(x, 0.0f) + log1pf(__expf(-fabsf(x)));   // stable softplus
      rowloss[rowBase + wave * 16 + half * 8 + r] = loss;
    }
  }
}

// ---------------- kernel 3: deterministic fixed-order mean reduction ----------------
__global__ void __launch_bounds__(256)
reduce_kernel(const float* __restrict__ rowloss, float* __restrict__ out) {
  __shared__ float s[256];
  const int tid = threadIdx.x;
  float acc = 0.f;
#pragma unroll
  for (int i = 0; i < N_ROWS / 256; ++i) acc += rowloss[tid + i * 256];
  s[tid] = acc;
  __syncthreads();
  for (int off = 128; off > 0; off >>= 1) {
    if (tid < off) s[tid] += s[tid + off];
    __syncthreads();
  }
  if (tid == 0) out[0] = s[0] * (1.0f / (float)N_ROWS);
}

extern "C" void kernel_launch(void* const* d_in, const int* in_sizes, int n_in,
                              void* d_out, int out_size, void* d_ws, size_t ws_size,
                              hipStream_t stream) {
  const float* X  = (const float*)d_in[0];   // [8192, 512] fp32
  const int*   tg = (const int*)d_in[1];     // [8192]
  float* sq      = (float*)d_ws;             // [8192]
  float* rowloss = sq + N_ROWS;              // [8192]
  float* out     = (float*)d_out;

  sqnorm_kernel<<<N_ROWS / 8, 256, 0, stream>>>(X, sq);

  size_t shmem = (size_t)(BLK_ROWS * LDS_STRIDE + 2 * CHUNK * LDS_STRIDE) * 4;
  triplet_main_kernel<<<N_ROWS / BLK_ROWS, THREADS, shmem, stream>>>(X, tg, sq, rowloss);

  reduce_kernel<<<1, 256, 0, stream>>>(rowloss, out);
}